// LSTMModel_42382737277070
// MI455X (gfx1250) — compile-verified
//
#include <hip/hip_runtime.h>

typedef float v2f __attribute__((ext_vector_type(2)));
typedef float v8f __attribute__((ext_vector_type(8)));

#define T_STEPS 512
#define D_IN    5
#define NP      208   // padded 4*H (200 -> 13 tiles of 16)
#define KP      52    // padded H (50 -> 13 k-steps of 4)
#define NT      13

// ws layout (v2f units unless noted):
//   P_ih0 : [4][208]   k-pair-interleaved transpose of W_ih0 (200x5, K padded to 8)
//   P_hh0 : [26][208]  of W_hh0 (200x50, K padded to 52)
//   P_ih1 : [26][208]  of W_ih1
//   P_hh1 : [26][208]  of W_hh1
//   bias0 : [208] float  (b_ih0 + b_hh0, zero padded)
//   bias1 : [208] float

__device__ __forceinline__ float fsig(float x) {
  float e = __builtin_amdgcn_exp2f(-1.442695041f * x);
  return __builtin_amdgcn_rcpf(1.0f + e);
}
__device__ __forceinline__ float ftanh_fast(float x) {
  float e = __builtin_amdgcn_exp2f(2.885390082f * x);   // e^{2x}
  return 1.0f - 2.0f * __builtin_amdgcn_rcpf(1.0f + e);
}

__device__ __forceinline__ void fill_pairs(v2f* P, const float* W, int Kin, int idx) {
  int p = idx / NP, n = idx % NP;
  int k = 2 * p;
  float a = (n < 200 && k     < Kin) ? W[n * Kin + k]     : 0.0f;
  float b = (n < 200 && (k+1) < Kin) ? W[n * Kin + k + 1] : 0.0f;
  v2f v; v.x = a; v.y = b;
  P[idx] = v;
}

__global__ void lstm_prep(const float* __restrict__ Wih0, const float* __restrict__ Whh0,
                          const float* __restrict__ bih0, const float* __restrict__ bhh0,
                          const float* __restrict__ Wih1, const float* __restrict__ Whh1,
                          const float* __restrict__ bih1, const float* __restrict__ bhh1,
                          float* __restrict__ ws)
{
  v2f* P_ih0 = (v2f*)ws;
  v2f* P_hh0 = P_ih0 + 4 * NP;
  v2f* P_ih1 = P_hh0 + 26 * NP;
  v2f* P_hh1 = P_ih1 + 26 * NP;
  float* b0  = (float*)(P_hh1 + 26 * NP);
  float* b1  = b0 + NP;

  int tid = blockIdx.x * blockDim.x + threadIdx.x;
  int stride = gridDim.x * blockDim.x;
  for (int i = tid; i < 4 * NP; i += stride)  fill_pairs(P_ih0, Wih0, D_IN, i);
  for (int i = tid; i < 26 * NP; i += stride) fill_pairs(P_hh0, Whh0, 50, i);
  for (int i = tid; i < 26 * NP; i += stride) fill_pairs(P_ih1, Wih1, 50, i);
  for (int i = tid; i < 26 * NP; i += stride) fill_pairs(P_hh1, Whh1, 50, i);
  for (int i = tid; i < NP; i += stride) {
    b0[i] = (i < 200) ? (bih0[i] + bhh0[i]) : 0.0f;
    b1[i] = (i < 200) ? (bih1[i] + bhh1[i]) : 0.0f;
  }
}

#define WMMA_F32(A, B, C) \
  __builtin_amdgcn_wmma_f32_16x16x4_f32(false, (A), false, (B), (short)0, (C), false, false)

__global__ __launch_bounds__(64) void lstm_fused(
    const float* __restrict__ x,     // (4096, 512, 5)
    const float* __restrict__ ws,
    const float* __restrict__ Wlin,  // (1, 50)
    const float* __restrict__ blin,  // (1,)
    float* __restrict__ out)         // (4096, 1)
{
  __shared__ __align__(16) float sG [2][16 * NP];
  __shared__ __align__(16) float sH1[2][16 * KP];
  __shared__ __align__(16) float sC1[2][16 * KP];
  __shared__ __align__(16) float sH2[2][16 * KP];
  __shared__ __align__(16) float sC2[2][16 * KP];

  const int tid  = threadIdx.x;
  const int w    = tid >> 5;       // wave id within block (0..1)
  const int lane = tid & 31;
  const int lmod = lane & 15;
  const int lhi  = lane >> 4;      // 0 or 1 (half-wave)
  const int b0r  = blockIdx.x * 32 + w * 16;

  float* G  = sG[w];
  float* H1 = sH1[w]; float* C1 = sC1[w];
  float* H2 = sH2[w]; float* C2 = sC2[w];

  const v2f* P_ih0 = (const v2f*)ws;
  const v2f* P_hh0 = P_ih0 + 4 * NP;
  const v2f* P_ih1 = P_hh0 + 26 * NP;
  const v2f* P_hh1 = P_ih1 + 26 * NP;
  const float* bias0 = (const float*)(P_hh1 + 26 * NP);
  const float* bias1 = bias0 + NP;

  // zero h/c state (padding columns 50..51 must stay zero forever)
  for (int i = lane; i < 16 * KP; i += 32) {
    H1[i] = 0.0f; C1[i] = 0.0f; H2[i] = 0.0f; C2[i] = 0.0f;
  }
  __builtin_amdgcn_fence(__ATOMIC_ACQ_REL, "workgroup");

  // per-lane bias values for accumulator init: column N = n*16 + lmod,
  // identical for all 8 rows of a C fragment and for both half-waves.
  float bb0[NT], bb1[NT];
  #pragma unroll
  for (int n = 0; n < NT; ++n) {
    bb0[n] = bias0[n * 16 + lmod];
    bb1[n] = bias1[n * 16 + lmod];
  }

  const float* xrow = x + (size_t)(b0r + lmod) * (T_STEPS * D_IN);
  const int   hb    = lhi * 25;                 // this lane's h-range start
  const float mask4 = (lhi == 0) ? 1.0f : 0.0f; // K=4 valid only in low half

  for (int t = 0; t < T_STEPS; ++t) {
    // ================= layer 0: gates = bias + x_t @ Wih0^T + h1 @ Whh0^T =========
    v8f acc[NT];
    #pragma unroll
    for (int n = 0; n < NT; ++n) {
      float b = bb0[n];
      acc[n] = (v8f){b, b, b, b, b, b, b, b};
    }

    const float* xt = xrow + t * D_IN;
    v2f a0; a0.x = xt[lhi * 2]; a0.y = xt[lhi * 2 + 1];  // K = 0..3
    v2f a1; a1.x = mask4 * xt[4]; a1.y = 0.0f;           // K = 4 (5..7 pad)

    #pragma unroll
    for (int n = 0; n < NT; ++n) {
      v2f b = P_ih0[(0 + lhi) * NP + n * 16 + lmod];
      acc[n] = WMMA_F32(a0, b, acc[n]);
    }
    #pragma unroll
    for (int n = 0; n < NT; ++n) {
      v2f b = P_ih0[(2 + lhi) * NP + n * 16 + lmod];
      acc[n] = WMMA_F32(a1, b, acc[n]);
    }
    for (int k0 = 0; k0 < KP; k0 += 4) {
      v2f a = *(const v2f*)&H1[lmod * KP + k0 + lhi * 2];
      const v2f* bp = P_hh0 + ((k0 >> 1) + lhi) * NP + lmod;
      #pragma unroll
      for (int n = 0; n < NT; ++n) {
        v2f b = bp[n * 16];
        acc[n] = WMMA_F32(a, b, acc[n]);
      }
    }
    #pragma unroll
    for (int n = 0; n < NT; ++n) {
      #pragma unroll
      for (int r = 0; r < 8; ++r)
        G[(r + lhi * 8) * NP + n * 16 + lmod] = acc[n][r];
    }
    __builtin_amdgcn_fence(__ATOMIC_ACQ_REL, "workgroup");

    // elementwise layer 0 (bias already folded into gates)
    for (int j = 0; j < 25; ++j) {
      int h = hb + j;
      float gi = G[lmod * NP + h];
      float gf = G[lmod * NP + 50 + h];
      float gg = G[lmod * NP + 100 + h];
      float go = G[lmod * NP + 150 + h];
      float c  = fsig(gf) * C1[lmod * KP + h] + fsig(gi) * ftanh_fast(gg);
      C1[lmod * KP + h] = c;
      H1[lmod * KP + h] = fsig(go) * ftanh_fast(c);
    }
    __builtin_amdgcn_fence(__ATOMIC_ACQ_REL, "workgroup");

    // ================= layer 1: gates = bias + h1_t @ Wih1^T + h2 @ Whh1^T ========
    #pragma unroll
    for (int n = 0; n < NT; ++n) {
      float b = bb1[n];
      acc[n] = (v8f){b, b, b, b, b, b, b, b};
    }

    for (int k0 = 0; k0 < KP; k0 += 4) {
      v2f a = *(const v2f*)&H1[lmod * KP + k0 + lhi * 2];
      const v2f* bp = P_ih1 + ((k0 >> 1) + lhi) * NP + lmod;
      #pragma unroll
      for (int n = 0; n < NT; ++n) {
        v2f b = bp[n * 16];
        acc[n] = WMMA_F32(a, b, acc[n]);
      }
    }
    for (int k0 = 0; k0 < KP; k0 += 4) {
      v2f a = *(const v2f*)&H2[lmod * KP + k0 + lhi * 2];
      const v2f* bp = P_hh1 + ((k0 >> 1) + lhi) * NP + lmod;
      #pragma unroll
      for (int n = 0; n < NT; ++n) {
        v2f b = bp[n * 16];
        acc[n] = WMMA_F32(a, b, acc[n]);
      }
    }
    #pragma unroll
    for (int n = 0; n < NT; ++n) {
      #pragma unroll
      for (int r = 0; r < 8; ++r)
        G[(r + lhi * 8) * NP + n * 16 + lmod] = acc[n][r];
    }
    __builtin_amdgcn_fence(__ATOMIC_ACQ_REL, "workgroup");

    // elementwise layer 1
    for (int j = 0; j < 25; ++j) {
      int h = hb + j;
      float gi = G[lmod * NP + h];
      float gf = G[lmod * NP + 50 + h];
      float gg = G[lmod * NP + 100 + h];
      float go = G[lmod * NP + 150 + h];
      float c  = fsig(gf) * C2[lmod * KP + h] + fsig(gi) * ftanh_fast(gg);
      C2[lmod * KP + h] = c;
      H2[lmod * KP + h] = fsig(go) * ftanh_fast(c);
    }
    __builtin_amdgcn_fence(__ATOMIC_ACQ_REL, "workgroup");
  }

  // final linear: out[b] = h2_last[b,:] . W_lin + b_lin
  if (lane < 16) {
    float dot = blin[0];
    for (int h = 0; h < 50; ++h)
      dot += H2[lmod * KP + h] * Wlin[h];
    out[b0r + lmod] = dot;
  }
}

extern "C" void kernel_launch(void* const* d_in, const int* in_sizes, int n_in,
                              void* d_out, int out_size, void* d_ws, size_t ws_size,
                              hipStream_t stream) {
  const float* x    = (const float*)d_in[0];
  const float* Wih0 = (const float*)d_in[1];
  const float* Whh0 = (const float*)d_in[2];
  const float* bih0 = (const float*)d_in[3];
  const float* bhh0 = (const float*)d_in[4];
  const float* Wih1 = (const float*)d_in[5];
  const float* Whh1 = (const float*)d_in[6];
  const float* bih1 = (const float*)d_in[7];
  const float* bhh1 = (const float*)d_in[8];
  const float* Wlin = (const float*)d_in[9];
  const float* blin = (const float*)d_in[10];
  float* ws  = (float*)d_ws;
  float* out = (float*)d_out;

  lstm_prep<<<dim3(64), dim3(256), 0, stream>>>(Wih0, Whh0, bih0, bhh0,
                                                Wih1, Whh1, bih1, bhh1, ws);
  // 4096 rows / (2 waves * 16 rows) = 128 blocks of 64 threads (2 waves)
  lstm_fused<<<dim3(128), dim3(64), 0, stream>>>(x, ws, Wlin, blin, out);
}